// SinkhornOrderingNet_1417339208329
// MI455X (gfx1250) — compile-verified
//
#include <hip/hip_runtime.h>
#include <hip/hip_bf16.h>
#include <math.h>

#define NN 8192
#define DD 512
#define HH 256

static constexpr float kStep   = 2.0f / 8191.0f;   // linspace(-1,1,N) step
static constexpr float kTauInv = 10.0f;            // 1/TAU
static constexpr float kClip   = -50.0f;
static constexpr float kEps    = 1e-6f;

typedef __attribute__((ext_vector_type(2))) float f32x2;
typedef __attribute__((ext_vector_type(8))) float f32x8;

// ---------------------------------------------------------------------------
// init small vectors
__global__ void init_vec_k(float* __restrict__ c, float* __restrict__ r) {
  int i = blockIdx.x * blockDim.x + threadIdx.x;
  if (i < NN) { c[i] = 0.0f; r[i] = 0.0f; }
}

// ---------------------------------------------------------------------------
// fp32 WMMA GEMM:  Cout[N,HH] = act(A[N,K] @ B[K,HH] + bias)
// One wave per 16x16 output tile, V_WMMA_F32_16X16X4_F32, K-loop of 4.
// A layout per lane: m = lane&15, a = {A[m][k0+kk], A[m][k0+kk+1]}, kk=2*(lane>>4)
// B layout per lane: n = lane&15, b = {B[k0+kk][n], B[k0+kk+1][n]}
// D layout: acc[v] -> row = v + 8*(lane>>4), col = lane&15
template<int K, bool GELU>
__global__ __launch_bounds__(256) void wmma_gemm_k(const float* __restrict__ A,
                                                   const float* __restrict__ B,
                                                   const float* __restrict__ bias,
                                                   float* __restrict__ Cout) {
  const int lane  = threadIdx.x & 31;
  const int wave  = threadIdx.x >> 5;
  const int tileM = blockIdx.y * 8 + wave;
  const int tileN = blockIdx.x;
  const int mh    = lane & 15;
  const int kk    = (lane >> 4) << 1;

  const float* Arow = A + (size_t)(tileM * 16 + mh) * K;
  const float* Bp   = B + tileN * 16 + mh;

  f32x8 acc = {};
  for (int k0 = 0; k0 < K; k0 += 4) {
    f32x2 a, b;
    a.x = Arow[k0 + kk];
    a.y = Arow[k0 + kk + 1];
    b.x = Bp[(size_t)(k0 + kk) * HH];
    b.y = Bp[(size_t)(k0 + kk + 1) * HH];
    acc = __builtin_amdgcn_wmma_f32_16x16x4_f32(false, a, false, b,
                                                (short)0, acc, false, false);
  }

  const int   col     = tileN * 16 + mh;
  const int   rowBase = tileM * 16 + ((lane >> 4) << 3);
  const float bv      = bias[col];
#pragma unroll
  for (int v = 0; v < 8; ++v) {
    float x = acc[v] + bv;
    if (GELU) x = 0.5f * x * (1.0f + erff(x * 0.70710678118654752f));
    Cout[(size_t)(rowBase + v) * HH + col] = x;
  }
}

// ---------------------------------------------------------------------------
// s0[i] = dot(E[i,:], w) + b  (one 256-thread block per row, H == 256)
__global__ __launch_bounds__(HH) void score_k(const float* __restrict__ E,
                                              const float* __restrict__ w,
                                              const float* __restrict__ b,
                                              float* __restrict__ s0) {
  __shared__ float red[HH];
  const int i = blockIdx.x;
  const int t = threadIdx.x;
  red[t] = E[(size_t)i * HH + t] * w[t];
  __syncthreads();
  for (int st = HH / 2; st > 0; st >>= 1) {
    if (t < st) red[t] += red[t + st];
    __syncthreads();
  }
  if (t == 0) s0[i] = red[0] + b[0];
}

// s[i] = s0[i] - mean(s0); also emits raw_scores output
__global__ __launch_bounds__(1024) void mean_sub_k(const float* __restrict__ s0,
                                                   float* __restrict__ s,
                                                   float* __restrict__ raw) {
  __shared__ float red[1024];
  const int t = threadIdx.x;
  float a = 0.0f;
  for (int j = t; j < NN; j += 1024) a += s0[j];
  red[t] = a;
  __syncthreads();
  for (int st = 512; st > 0; st >>= 1) {
    if (t < st) red[t] += red[t + st];
    __syncthreads();
  }
  const float mean = red[0] * (1.0f / NN);
  for (int j = t; j < NN; j += 1024) {
    const float v = s0[j] - mean;
    s[j]   = v;
    raw[j] = v;
  }
}

// ---------------------------------------------------------------------------
// L(i,j) = max(-10*(s_i - p_j)^2, -50),  p_j = -1 + j*kStep
__device__ __forceinline__ float Lval(float si, float pj) {
  const float d = si - pj;
  return fmaxf(-kTauInv * d * d, kClip);
}

// r[i] = -LSE_j( L(i,j) + c[j] ), 8 rows per block, two-pass max/sum
__global__ __launch_bounds__(256) void row_step_k(const float* __restrict__ s,
                                                  const float* __restrict__ c,
                                                  float* __restrict__ r) {
  __shared__ float part[8][256];
  const int t  = threadIdx.x;
  const int i0 = blockIdx.x * 8;

  float si[8];
#pragma unroll
  for (int q = 0; q < 8; ++q) si[q] = s[i0 + q];

  float mx[8];
#pragma unroll
  for (int q = 0; q < 8; ++q) mx[q] = -3.0e38f;

  for (int j = t; j < NN; j += 256) {
    const float p  = fmaf((float)j, kStep, -1.0f);
    const float cj = c[j];
#pragma unroll
    for (int q = 0; q < 8; ++q) mx[q] = fmaxf(mx[q], Lval(si[q], p) + cj);
  }
#pragma unroll
  for (int q = 0; q < 8; ++q) part[q][t] = mx[q];
  __syncthreads();
  for (int st = 128; st > 0; st >>= 1) {
    if (t < st) {
#pragma unroll
      for (int q = 0; q < 8; ++q) part[q][t] = fmaxf(part[q][t], part[q][t + st]);
    }
    __syncthreads();
  }
  float M[8];
#pragma unroll
  for (int q = 0; q < 8; ++q) M[q] = part[q][0];
  __syncthreads();

  float sm[8] = {0, 0, 0, 0, 0, 0, 0, 0};
  for (int j = t; j < NN; j += 256) {
    const float p  = fmaf((float)j, kStep, -1.0f);
    const float cj = c[j];
#pragma unroll
    for (int q = 0; q < 8; ++q) sm[q] += __expf(Lval(si[q], p) + cj - M[q]);
  }
#pragma unroll
  for (int q = 0; q < 8; ++q) part[q][t] = sm[q];
  __syncthreads();
  for (int st = 128; st > 0; st >>= 1) {
    if (t < st) {
#pragma unroll
      for (int q = 0; q < 8; ++q) part[q][t] += part[q][t + st];
    }
    __syncthreads();
  }
  if (t < 8) r[i0 + t] = -(M[t] + __logf(part[t][0]));
}

// c[j] = -LSE_i( L(i,j) + r[i] ), 8 columns per block
__global__ __launch_bounds__(256) void col_step_k(const float* __restrict__ s,
                                                  const float* __restrict__ r,
                                                  float* __restrict__ c) {
  __shared__ float part[8][256];
  const int t  = threadIdx.x;
  const int j0 = blockIdx.x * 8;

  float pj[8];
#pragma unroll
  for (int q = 0; q < 8; ++q) pj[q] = fmaf((float)(j0 + q), kStep, -1.0f);

  float mx[8];
#pragma unroll
  for (int q = 0; q < 8; ++q) mx[q] = -3.0e38f;

  for (int i = t; i < NN; i += 256) {
    const float si = s[i];
    const float ri = r[i];
#pragma unroll
    for (int q = 0; q < 8; ++q) mx[q] = fmaxf(mx[q], Lval(si, pj[q]) + ri);
  }
#pragma unroll
  for (int q = 0; q < 8; ++q) part[q][t] = mx[q];
  __syncthreads();
  for (int st = 128; st > 0; st >>= 1) {
    if (t < st) {
#pragma unroll
      for (int q = 0; q < 8; ++q) part[q][t] = fmaxf(part[q][t], part[q][t + st]);
    }
    __syncthreads();
  }
  float M[8];
#pragma unroll
  for (int q = 0; q < 8; ++q) M[q] = part[q][0];
  __syncthreads();

  float sm[8] = {0, 0, 0, 0, 0, 0, 0, 0};
  for (int i = t; i < NN; i += 256) {
    const float si = s[i];
    const float ri = r[i];
#pragma unroll
    for (int q = 0; q < 8; ++q) sm[q] += __expf(Lval(si, pj[q]) + ri - M[q]);
  }
#pragma unroll
  for (int q = 0; q < 8; ++q) part[q][t] = sm[q];
  __syncthreads();
  for (int st = 128; st > 0; st >>= 1) {
    if (t < st) {
#pragma unroll
      for (int q = 0; q < 8; ++q) part[q][t] += part[q][t + st];
    }
    __syncthreads();
  }
  if (t < 8) c[j0 + t] = -(M[t] + __logf(part[t][0]));
}

// rowInv[i] = 1 / (sum_j exp(L + r_i + c_j) + eps)
__global__ __launch_bounds__(256) void row_sum_k(const float* __restrict__ s,
                                                 const float* __restrict__ r,
                                                 const float* __restrict__ c,
                                                 float* __restrict__ rowInv) {
  __shared__ float part[8][256];
  const int t  = threadIdx.x;
  const int i0 = blockIdx.x * 8;
  float si[8], ri[8];
#pragma unroll
  for (int q = 0; q < 8; ++q) { si[q] = s[i0 + q]; ri[q] = r[i0 + q]; }
  float sm[8] = {0, 0, 0, 0, 0, 0, 0, 0};
  for (int j = t; j < NN; j += 256) {
    const float p  = fmaf((float)j, kStep, -1.0f);
    const float cj = c[j];
#pragma unroll
    for (int q = 0; q < 8; ++q) sm[q] += __expf(Lval(si[q], p) + ri[q] + cj);
  }
#pragma unroll
  for (int q = 0; q < 8; ++q) part[q][t] = sm[q];
  __syncthreads();
  for (int st = 128; st > 0; st >>= 1) {
    if (t < st) {
#pragma unroll
      for (int q = 0; q < 8; ++q) part[q][t] += part[q][t + st];
    }
    __syncthreads();
  }
  if (t < 8) rowInv[i0 + t] = 1.0f / (part[t][0] + kEps);
}

// colInv[j] = 1/(C_j+eps) with C_j = sum_i m1(i,j);  er[j] = (sum_i i*m1(i,j)) * colInv[j]
__global__ __launch_bounds__(256) void col_sum_k(const float* __restrict__ s,
                                                 const float* __restrict__ r,
                                                 const float* __restrict__ c,
                                                 const float* __restrict__ rowInv,
                                                 float* __restrict__ colInv,
                                                 float* __restrict__ er) {
  __shared__ float part[8][256];
  const int t  = threadIdx.x;
  const int j0 = blockIdx.x * 8;
  float pj[8], cj[8];
#pragma unroll
  for (int q = 0; q < 8; ++q) {
    pj[q] = fmaf((float)(j0 + q), kStep, -1.0f);
    cj[q] = c[j0 + q];
  }
  float sC[8] = {0, 0, 0, 0, 0, 0, 0, 0};
  float sW[8] = {0, 0, 0, 0, 0, 0, 0, 0};
  for (int i = t; i < NN; i += 256) {
    const float si = s[i];
    const float ri = r[i];
    const float rv = rowInv[i];
    const float fi = (float)i;
#pragma unroll
    for (int q = 0; q < 8; ++q) {
      const float e = __expf(Lval(si, pj[q]) + ri + cj[q]) * rv;
      sC[q] += e;
      sW[q] += e * fi;
    }
  }
#pragma unroll
  for (int q = 0; q < 8; ++q) part[q][t] = sC[q];
  __syncthreads();
  for (int st = 128; st > 0; st >>= 1) {
    if (t < st) {
#pragma unroll
      for (int q = 0; q < 8; ++q) part[q][t] += part[q][t + st];
    }
    __syncthreads();
  }
  float Cred[8];
#pragma unroll
  for (int q = 0; q < 8; ++q) Cred[q] = part[q][0];
  __syncthreads();
#pragma unroll
  for (int q = 0; q < 8; ++q) part[q][t] = sW[q];
  __syncthreads();
  for (int st = 128; st > 0; st >>= 1) {
    if (t < st) {
#pragma unroll
      for (int q = 0; q < 8; ++q) part[q][t] += part[q][t + st];
    }
    __syncthreads();
  }
  if (t < 8) {
    const float ci = 1.0f / (Cred[t] + kEps);
    colInv[j0 + t] = ci;
    er[j0 + t]     = part[t][0] * ci;
  }
}

// write perm_matrix and logits (float4 stores; HBM-bound: ~537 MB)
__global__ __launch_bounds__(256) void final_write_k(const float* __restrict__ s,
                                                     const float* __restrict__ r,
                                                     const float* __restrict__ c,
                                                     const float* __restrict__ rowInv,
                                                     const float* __restrict__ colInv,
                                                     float* __restrict__ perm,
                                                     float* __restrict__ logits) {
  const int i  = blockIdx.y;
  const int j0 = (blockIdx.x * 256 + threadIdx.x) * 4;
  const float si = s[i];
  const float ri = r[i];
  const float rv = rowInv[i];
  const float4 cj4 = *(const float4*)(c + j0);
  const float4 cv4 = *(const float4*)(colInv + j0);
  const float cj[4] = {cj4.x, cj4.y, cj4.z, cj4.w};
  const float cv[4] = {cv4.x, cv4.y, cv4.z, cv4.w};
  float lg[4], pm[4];
#pragma unroll
  for (int q = 0; q < 4; ++q) {
    const float p = fmaf((float)(j0 + q), kStep, -1.0f);
    const float d = si - p;
    const float L = -kTauInv * d * d;
    lg[q] = L;
    pm[q] = __expf(fmaxf(L, kClip) + ri + cj[q]) * rv * cv[q];
  }
  const size_t base = (size_t)i * NN + j0;
  *(float4*)(logits + base) = make_float4(lg[0], lg[1], lg[2], lg[3]);
  *(float4*)(perm + base)   = make_float4(pm[0], pm[1], pm[2], pm[3]);
}

// ---------------------------------------------------------------------------
extern "C" void kernel_launch(void* const* d_in, const int* in_sizes, int n_in,
                              void* d_out, int out_size, void* d_ws, size_t ws_size,
                              hipStream_t stream) {
  const float* comp = (const float*)d_in[0];  // [N,D]
  const float* W1   = (const float*)d_in[1];  // [D,H]
  const float* b1   = (const float*)d_in[2];  // [H]
  const float* W2   = (const float*)d_in[3];  // [H,H]
  const float* b2   = (const float*)d_in[4];  // [H]
  const float* wsc  = (const float*)d_in[5];  // [H]
  const float* bsc  = (const float*)d_in[6];  // [1]

  float* out    = (float*)d_out;
  float* perm   = out;                                  // [N,N]
  float* logits = out + (size_t)NN * NN;                // [N,N]
  float* er     = out + 2ull * NN * NN;                 // [N]
  float* raw    = er + NN;                              // [N]

  // stage the two 8MB MLP intermediates inside the perm region (overwritten last)
  float* H1 = perm;                       // [N,H]
  float* E  = perm + (size_t)NN * HH;     // [N,H]

  float* ws     = (float*)d_ws;           // needs 6*N floats = 192 KB
  float* s0     = ws;
  float* s      = ws + 1 * NN;
  float* r      = ws + 2 * NN;
  float* c      = ws + 3 * NN;
  float* rowInv = ws + 4 * NN;
  float* colInv = ws + 5 * NN;

  init_vec_k<<<NN / 256, 256, 0, stream>>>(c, r);

  // MLP: WMMA fp32 GEMMs
  wmma_gemm_k<DD, true ><<<dim3(HH / 16, NN / 16 / 8), 256, 0, stream>>>(comp, W1, b1, H1);
  wmma_gemm_k<HH, false><<<dim3(HH / 16, NN / 16 / 8), 256, 0, stream>>>(H1, W2, b2, E);
  score_k<<<NN, HH, 0, stream>>>(E, wsc, bsc, s0);
  mean_sub_k<<<1, 1024, 0, stream>>>(s0, s, raw);

  // matrix-free Sinkhorn on the separable form L(i,j) + r[i] + c[j]
  for (int it = 0; it < 30; ++it) {
    row_step_k<<<NN / 8, 256, 0, stream>>>(s, c, r);
    col_step_k<<<NN / 8, 256, 0, stream>>>(s, r, c);
  }
  row_sum_k<<<NN / 8, 256, 0, stream>>>(s, r, c, rowInv);
  col_sum_k<<<NN / 8, 256, 0, stream>>>(s, r, c, rowInv, colInv, er);
  final_write_k<<<dim3(NN / 1024, NN), 256, 0, stream>>>(s, r, c, rowInv, colInv,
                                                         perm, logits);
}